// GraphModule_59012850647682
// MI455X (gfx1250) — compile-verified
//
#include <hip/hip_runtime.h>

#define N_NODES 100000
#define N_EDGES 640000
#define D       128
#define ROWS_PER_BLOCK 80      // 100000 / 80 = 1250 blocks, exact
#define ROW_TILES 5            // 80 / 16
#define LDS_STRIDE 132         // pad 128 -> 132 for conflict-free column reads

typedef float v2f __attribute__((ext_vector_type(2)));
typedef float v8f __attribute__((ext_vector_type(8)));

// ---------------------------------------------------------------------------
// GEMM: out[m][n] = sum_k relu?(in[m][k]) * w[n][k]     (h @ W.T)
// One block = 80 rows; 8 waves, each wave owns a 16-col tile, 5 row tiles.
// f32 WMMA 16x16x4: A 16x4 (v2f/lane), B 4x16 (v2f/lane), C/D 16x16 (v8f/lane)
// ---------------------------------------------------------------------------
__global__ __launch_bounds__(256) void k_gemm(const float* __restrict__ in,
                                              const float* __restrict__ w,
                                              float* __restrict__ out,
                                              int apply_relu) {
    __shared__ float sH[ROWS_PER_BLOCK * LDS_STRIDE];   // ~42 KB
    const int tid  = threadIdx.x;
    const int row0 = blockIdx.x * ROWS_PER_BLOCK;

    // Stage 80x128 input tile into LDS (coalesced), fold ReLU of prev layer.
    for (int it = 0; it < (ROWS_PER_BLOCK * D) / 256; ++it) {
        int idx = it * 256 + tid;
        int r = idx >> 7;          // row in tile
        int k = idx & 127;         // k index
        float v = in[(size_t)(row0 + r) * D + k];
        if (apply_relu) v = fmaxf(v, 0.0f);
        sH[r * LDS_STRIDE + k] = v;
    }
    __syncthreads();

    const int lane = tid & 31;
    const int wv   = tid >> 5;         // wave id 0..7
    const int col0 = wv * 16;          // output column tile
    const int half = lane >> 4;        // 0: K+0/K+1 ; 1: K+2/K+3
    const int mn   = lane & 15;        // M (for A) / N (for B,C,D)

    v8f acc[ROW_TILES];
#pragma unroll
    for (int rt = 0; rt < ROW_TILES; ++rt)
        acc[rt] = (v8f){0.f,0.f,0.f,0.f,0.f,0.f,0.f,0.f};

    for (int k0 = 0; k0 < D; k0 += 4) {
        const int kA = k0 + 2 * half;
        // B fragment: B[k][n] = W[col0+n][k]  (W cached in WGP$/L2, 64 KB)
        v2f b = *(const v2f*)(w + (size_t)(col0 + mn) * D + kA);
#pragma unroll
        for (int rt = 0; rt < ROW_TILES; ++rt) {
            // A fragment from LDS (row mn of this 16-row tile, K pair kA)
            v2f a = *(const v2f*)(sH + (rt * 16 + mn) * LDS_STRIDE + kA);
            acc[rt] = __builtin_amdgcn_wmma_f32_16x16x4_f32(
                false, a, false, b, (short)0, acc[rt], false, false);
        }
    }

    // C/D layout: VGPR r holds row (r + 8*half), lane mn is column mn.
#pragma unroll
    for (int rt = 0; rt < ROW_TILES; ++rt) {
#pragma unroll
        for (int r = 0; r < 8; ++r) {
            out[(size_t)(row0 + rt * 16 + r + 8 * half) * D + col0 + mn] = acc[rt][r];
        }
    }
}

// ---------------------------------------------------------------------------
// Degree / normalization (computed once per launch; identical every layer)
// ---------------------------------------------------------------------------
__global__ void k_deg_init(float* __restrict__ deg) {
    int i = blockIdx.x * blockDim.x + threadIdx.x;
    if (i < N_NODES) deg[i] = 1.0f;                     // self loop
}

__global__ void k_deg_acc(const int* __restrict__ src, const int* __restrict__ dst,
                          float* __restrict__ deg) {
    int e = blockIdx.x * blockDim.x + threadIdx.x;
    if (e < N_EDGES) {
        int s = src[e], t = dst[e];
        if (s != t) unsafeAtomicAdd(&deg[t], 1.0f);
    }
}

__global__ void k_dinv(float* __restrict__ deg) {
    int i = blockIdx.x * blockDim.x + threadIdx.x;
    if (i < N_NODES) deg[i] = rsqrtf(deg[i]);           // deg >= 1 always
}

__global__ void k_norm(const int* __restrict__ src, const int* __restrict__ dst,
                       const float* __restrict__ dinv, float* __restrict__ norm) {
    int e = blockIdx.x * blockDim.x + threadIdx.x;
    if (e < N_EDGES) {
        int s = src[e], t = dst[e];
        norm[e] = (s != t) ? dinv[s] * dinv[t] : 0.0f;
    }
}

// ---------------------------------------------------------------------------
// out[i][:] = dinv[i]^2 * hlin[i][:] + b[:]   (self-loop term + bias, init)
// one thread per float4
// ---------------------------------------------------------------------------
__global__ __launch_bounds__(256) void k_self_bias(const float* __restrict__ hlin,
                                                   const float* __restrict__ dinv,
                                                   const float* __restrict__ bias,
                                                   float* __restrict__ out) {
    int idx = blockIdx.x * blockDim.x + threadIdx.x;    // over N*D/4
    int i  = idx >> 5;                                  // node (D/4 == 32)
    int c4 = idx & 31;                                  // float4 column
    float di = dinv[i];
    float s  = di * di;
    float4 hv = ((const float4*)hlin)[idx];
    float4 bv = ((const float4*)bias)[c4];
    float4 ov;
    ov.x = fmaf(s, hv.x, bv.x);
    ov.y = fmaf(s, hv.y, bv.y);
    ov.z = fmaf(s, hv.z, bv.z);
    ov.w = fmaf(s, hv.w, bv.w);
    ((float4*)out)[idx] = ov;
}

// ---------------------------------------------------------------------------
// Edge scatter: out[dst[e]][:] += norm[e] * hlin[src[e]][:]
// One wave per edge: 32 lanes x float4 = full 128-float row, coalesced 512B.
// L2-resident (51.2 MB matrix), native global_atomic_add_f32.
// ---------------------------------------------------------------------------
__global__ __launch_bounds__(256) void k_edge(const int* __restrict__ src,
                                              const int* __restrict__ dst,
                                              const float* __restrict__ norm,
                                              const float* __restrict__ hlin,
                                              float* __restrict__ out) {
    int e    = blockIdx.x * 8 + (threadIdx.x >> 5);
    int lane = threadIdx.x & 31;
    int s = src[e], t = dst[e];
    if (s == t) return;                                  // norm == 0 edge
    float nm = norm[e];
    float4 v = ((const float4*)hlin)[(size_t)s * 32 + lane];
    float* o = out + (size_t)t * D + lane * 4;
    unsafeAtomicAdd(o + 0, nm * v.x);
    unsafeAtomicAdd(o + 1, nm * v.y);
    unsafeAtomicAdd(o + 2, nm * v.z);
    unsafeAtomicAdd(o + 3, nm * v.w);
}

// ---------------------------------------------------------------------------
extern "C" void kernel_launch(void* const* d_in, const int* in_sizes, int n_in,
                              void* d_out, int out_size, void* d_ws, size_t ws_size,
                              hipStream_t stream) {
    const float* x   = (const float*)d_in[0];
    const int*   ei  = (const int*)d_in[1];
    const int*   src = ei;                 // edge_index[0]
    const int*   dst = ei + N_EDGES;       // edge_index[1]
    const float* W[5];
    const float* B[5];
    for (int l = 0; l < 5; ++l) {
        W[l] = (const float*)d_in[2 + 2 * l];
        B[l] = (const float*)d_in[3 + 2 * l];
    }
    float* out = (float*)d_out;

    // Workspace layout (floats): dinv[N] | norm[E] | bufA[N*D] | bufB[N*D]
    float* ws   = (float*)d_ws;
    float* dinv = ws;
    float* nrm  = ws + N_NODES;
    float* bufA = ws + N_NODES + N_EDGES;              // hlin (post-GEMM)
    float* bufB = bufA + (size_t)N_NODES * D;          // h ping-pong

    // --- normalization (once; identical across layers) ---
    k_deg_init<<<(N_NODES + 255) / 256, 256, 0, stream>>>(dinv);
    k_deg_acc <<<(N_EDGES + 255) / 256, 256, 0, stream>>>(src, dst, dinv);
    k_dinv    <<<(N_NODES + 255) / 256, 256, 0, stream>>>(dinv);
    k_norm    <<<(N_EDGES + 255) / 256, 256, 0, stream>>>(src, dst, dinv, nrm);

    // --- 5 GCN layers ---
    const float* hin = x;
    for (int l = 0; l < 5; ++l) {
        // hlin = relu?(hin) @ W.T   (ReLU of previous layer folded in)
        k_gemm<<<N_NODES / ROWS_PER_BLOCK, 256, 0, stream>>>(
            hin, W[l], bufA, (l > 0) ? 1 : 0);
        float* hout = (l == 4) ? out : bufB;
        // hout = dinv^2 * hlin + bias  (init), then scatter-add edges
        k_self_bias<<<(N_NODES * (D / 4)) / 256, 256, 0, stream>>>(
            bufA, dinv, B[l], hout);
        k_edge<<<N_EDGES / 8, 256, 0, stream>>>(src, dst, nrm, bufA, hout);
        hin = hout;   // bufB consumed by next GEMM before being overwritten
    }
}